// RPE_9491877724456
// MI455X (gfx1250) — compile-verified
//
#include <hip/hip_runtime.h>

typedef __attribute__((ext_vector_type(2))) float v2f;
typedef __attribute__((ext_vector_type(8))) float v8f;

// Problem constants (match the reference)
constexpr int Bb = 4, Dd = 1024, Hh = 12, Tt = 32, Ss = 32, FD = 64;
constexpr int NUM_BUCKETS = 32;
constexpr int MAX_BUCKET  = 16;   // NUM_BUCKETS/2
constexpr int MIN_BUCKET  = -15;  // MAX - NUM_BUCKETS + 1

// One block per (b,h,t). 8 waves; each wave computes 8 tiles of 16(D) x 32(S)
// via V_WMMA_F32_16X16X4_F32, accumulating over K = FD = 64 in steps of 4.
//
// Memory policy (MI455X, 23.3 TB/s HBM, 192 MB L2):
//  - qk (402 MB) is streamed exactly once  -> non-temporal loads (TH=NT)
//  - out (201 MB) is written exactly once   -> non-temporal stores (TH=NT)
//  - lut (98 KB) is reused by all blocks   -> regular temporal loads, stays in L2
__global__ __launch_bounds__(256) void rpe_qk_wmma(
    const float* __restrict__ qk,     // [B, D, H, T, FD]
    const int*   __restrict__ pdist,  // [B, T, S]
    const float* __restrict__ lut,    // [NUM_BUCKETS, H, FD]
    float*       __restrict__ out)    // [B, D, H, T, S]
{
  // Gathered RPE matrix for this (b,t,h): R[s][f], padded row stride 66
  // (even -> 8-byte aligned float2 rows; stride-2 banks -> conflict free).
  __shared__ float Bs[Ss][FD + 2];

  const int bid = blockIdx.x;
  const int t = bid % Tt;
  const int h = (bid / Tt) % Hh;
  const int b = bid / (Tt * Hh);

  const int tid = threadIdx.x;

  // ---- Stage B matrix: gather lut rows by clamped/wrapped bucket index ----
  for (int i = tid; i < Ss * FD; i += 256) {
    const int s = i >> 6;          // FD == 64
    const int f = i & (FD - 1);
    int pd = pdist[(b * Tt + t) * Ss + s];
    pd = pd < MIN_BUCKET ? MIN_BUCKET : (pd > MAX_BUCKET ? MAX_BUCKET : pd);
    const int idx = pd < 0 ? pd + NUM_BUCKETS : pd;
    Bs[s][f] = lut[(idx * Hh + h) * FD + f];
  }
  __syncthreads();

  const int lane = tid & 31;
  const int lm   = lane & 15;   // row of A-tile / column of B-tile
  const int lh   = lane >> 4;   // selects K pair {0,1} vs {2,3} within K=4
  const int wave = tid >> 5;    // 0..7

  // Per-lane LDS base pointers for the two S-halves of the B matrix.
  // (Compiler hoists all 16 k-steps of both into VGPRs, reused by all m-tiles.)
  const float* bp0 = &Bs[lm][lh * 2];
  const float* bp1 = &Bs[lm + 16][lh * 2];

  // out flat index = b*D*H*T*S + d*(H*T*S) + h*T*S + t*S + s
  const long hts   = (long)Hh * Tt * Ss;                       // 12288
  const long oBase = (long)b * Dd * hts + (long)h * Tt * Ss + (long)t * Ss;

  for (int iter = 0; iter < 8; ++iter) {
    const int mbase = (wave + iter * 8) * 16;

    // qk flat index = (((b*D + d)*H + h)*T + t)*FD + f ; lane reads row mbase+lm
    const float* aRow =
        qk + ((((long)b * Dd + mbase + lm) * Hh + h) * Tt + t) * FD + lh * 2;

    v8f acc0 = {};
    v8f acc1 = {};
#pragma unroll
    for (int kb = 0; kb < FD / 4; ++kb) {
      // A is streamed once across the whole grid: non-temporal.
      const v2f a  = __builtin_nontemporal_load((const v2f*)(aRow + kb * 4));
      const v2f b0 = *(const v2f*)(bp0 + kb * 4);    // B[k][lm],     s = lm
      const v2f b1 = *(const v2f*)(bp1 + kb * 4);    // B[k][lm+16],  s = lm+16
      acc0 = __builtin_amdgcn_wmma_f32_16x16x4_f32(
          false, a, false, b0, (short)0, acc0, false, false);
      acc1 = __builtin_amdgcn_wmma_f32_16x16x4_f32(
          false, a, false, b1, (short)0, acc1, false, false);
    }

    // D-matrix layout: VGPR v, lanes 0-15 -> M=v, lanes 16-31 -> M=v+8; N=lm.
    // Each half-wave store is 16 lanes x 4 B = 64 B contiguous.
#pragma unroll
    for (int v = 0; v < 8; ++v) {
      const long d = mbase + v + lh * 8;
      float* o = out + oBase + d * hts + lm;
      __builtin_nontemporal_store(acc0[v], o);        // s = lm
      __builtin_nontemporal_store(acc1[v], o + 16);   // s = lm + 16
    }
  }
}

extern "C" void kernel_launch(void* const* d_in, const int* in_sizes, int n_in,
                              void* d_out, int out_size, void* d_ws, size_t ws_size,
                              hipStream_t stream) {
  const float* qk    = (const float*)d_in[0];
  const int*   pdist = (const int*)d_in[1];
  const float* lut   = (const float*)d_in[2];
  float*       out   = (float*)d_out;

  dim3 grid(Bb * Hh * Tt);   // 1536 blocks, one per (b,h,t)
  dim3 block(256);           // 8 wave32s
  hipLaunchKernelGGL(rpe_qk_wmma, grid, block, 0, stream,
                     qk, pdist, lut, out);
}